// SimDiVeQ_73839077753415
// MI455X (gfx1250) — compile-verified
//
#include <hip/hip_runtime.h>
#include <math.h>

// ---------------------------------------------------------------------------
// DiVeQ nearest-code search on MI455X (gfx1250).
//   N = 32768 rows, D = 64, K = 8192 codes.
//   Score GEMM (32768 x 8192 x 64, fp32) done with V_WMMA_F32_16X16X4_F32.
// ---------------------------------------------------------------------------

#define DIMD   64
#define KCODES 8192
#define NROWS  32768
#define NTILES (KCODES / 16)   // 512 code tiles of 16
#define EPSF   1e-5f

typedef float v2f __attribute__((ext_vector_type(2)));
typedef float v8f __attribute__((ext_vector_type(8)));

// ---------------------------------------------------------------------------
// Kernel A: codebook = frozen @ W^T ; pack as float2 pk[p][k] = (cb[k][2p], cb[k][2p+1])
// and cnorm[k] = ||cb[k]||^2.  8192 blocks x 64 threads — negligible cost.
// ---------------------------------------------------------------------------
__global__ __launch_bounds__(64) void diveq_prep(
    const float* __restrict__ frozen,   // [8192][64]
    const float* __restrict__ W,        // [64][64]  (out, in)
    float2* __restrict__ pk,            // [32][8192]
    float* __restrict__ cnorm,          // [8192]
    float* __restrict__ loss_out)
{
    const int k = blockIdx.x;      // code id
    const int t = threadIdx.x;     // output dim
    __shared__ float fr[DIMD];
    __shared__ float cb[DIMD];

    fr[t] = frozen[k * DIMD + t];
    __syncthreads();

    float acc = 0.f;
#pragma unroll 8
    for (int j = 0; j < DIMD; ++j)
        acc = fmaf(fr[j], W[t * DIMD + j], acc);   // cb[k][t] = sum_j frozen[k][j]*W[t][j]
    cb[t] = acc;
    __syncthreads();

    if (t < 32) {
        float2 p;
        p.x = cb[2 * t];
        p.y = cb[2 * t + 1];
        pk[t * KCODES + k] = p;
    }
    if (t == 0) {
        float s = 0.f;
#pragma unroll 8
        for (int j = 0; j < DIMD; ++j) s = fmaf(cb[j], cb[j], s);
        cnorm[k] = s;
        if (k == 0) loss_out[0] = 0.f;   // reference loss is constant 0
    }
}

// ---------------------------------------------------------------------------
// Kernel B: per-wave 16-row tile; loop over 512 code tiles with f32 WMMA.
// Block = 256 threads (8 waves) -> 128 rows/block, grid = 256 blocks.
// B tiles double-buffered in LDS, shared by all 8 waves.
// ---------------------------------------------------------------------------
__global__ __launch_bounds__(256) void diveq_main(
    const float* __restrict__ x,          // [32768][64]
    const unsigned char* __restrict__ mask,  // [32768] bool
    const float2* __restrict__ pk,        // [32][8192] packed codebook
    const float* __restrict__ cnorm,      // [8192]
    float* __restrict__ out_q,            // [32768][64]
    float* __restrict__ out_idx)          // [32768] (as float)
{
    __shared__ float  s_cnorm[KCODES];        // 32 KB
    __shared__ float2 s_btile[2][32 * 16];    // 2 x 4 KB double buffer

    const int tid   = threadIdx.x;
    const int wave  = tid >> 5;
    const int lane  = tid & 31;
    const int half  = lane >> 4;
    const int nlane = lane & 15;

    // Stage all ||c||^2 into LDS once.
    for (int i = tid; i < KCODES; i += 256) s_cnorm[i] = cnorm[i];

    // Load this wave's A tile (16 rows x 64) into registers.
    // ISA 16x4 f32 A layout: lane m = row (lane%16), VGPR v -> K = v + 2*(lane/16).
    const int row_base = (blockIdx.x * 8 + wave) * 16;
    const int arow     = row_base + nlane;
    const float2* __restrict__ x2 = (const float2*)x;

    v2f a[16];
#pragma unroll
    for (int c = 0; c < 16; ++c) {
        float2 t = x2[arow * 32 + c * 2 + half];   // dims (4c+2*half, 4c+2*half+1)
        a[c] = (v2f){t.x, t.y};
    }

    float minv[8];
    int   mini[8];
#pragma unroll
    for (int v = 0; v < 8; ++v) { minv[v] = __builtin_inff(); mini[v] = 0; }

    auto stage = [&](int buf, int nt) {
        // 512 float2 elements: element e -> p = e/16 (K pair), n = e%16 (code in tile)
        int e0 = tid, e1 = tid + 256;
        s_btile[buf][e0] = pk[(e0 >> 4) * KCODES + nt * 16 + (e0 & 15)];
        s_btile[buf][e1] = pk[(e1 >> 4) * KCODES + nt * 16 + (e1 & 15)];
    };

    stage(0, 0);
    __syncthreads();

    for (int nt = 0; nt < NTILES; ++nt) {
        const int buf = nt & 1;
        if (nt + 1 < NTILES) stage(buf ^ 1, nt + 1);

        v8f acc = {};
#pragma unroll
        for (int c = 0; c < 16; ++c) {
            // B layout (4x16): lane n = col (lane%16), VGPR v -> K = v + 2*(lane/16)
            float2 bb = s_btile[buf][(c * 2 + half) * 16 + nlane];
            v2f b = (v2f){bb.x, bb.y};
            acc = __builtin_amdgcn_wmma_f32_16x16x4_f32(
                false, a[c], false, b, (short)0, acc, false, false);
        }

        const int   code = nt * 16 + nlane;
        const float cn   = s_cnorm[code];
#pragma unroll
        for (int v = 0; v < 8; ++v) {
            float s = fmaf(-2.f, acc[v], cn);     // ||c||^2 - 2 x.c
            if (s < minv[v]) { minv[v] = s; mini[v] = code; }  // strict < keeps first min
        }
        __syncthreads();
    }

    // Reduce (min, argmin) across the 16 lanes of each half (xor masks stay in-half).
#pragma unroll
    for (int v = 0; v < 8; ++v) {
#pragma unroll
        for (int m = 1; m <= 8; m <<= 1) {
            float ov = __shfl_xor(minv[v], m, 32);
            int   oi = __shfl_xor(mini[v], m, 32);
            if (ov < minv[v] || (ov == minv[v] && oi < mini[v])) {
                minv[v] = ov; mini[v] = oi;
            }
        }
    }

    // Epilogue: gather winning code, DiVeQ forward, per row.
#pragma unroll
    for (int r = 0; r < 16; ++r) {
        const int v = r & 7, h = r >> 3;
        const int idx  = __shfl(mini[v], h * 16, 32);   // row r lives in half h, VGPR v
        const int grow = row_base + r;

        // lane holds dims (2*lane, 2*lane+1): pk[p=lane][idx] = (cb[idx][2p], cb[idx][2p+1])
        float2 cb2 = pk[lane * KCODES + idx];
        float2 xv  = x2[grow * 32 + lane];
        float d0 = cb2.x - xv.x;
        float d1 = cb2.y - xv.y;
        float ss = fmaf(d0, d0, d1 * d1);
#pragma unroll
        for (int m = 1; m <= 16; m <<= 1) ss += __shfl_xor(ss, m, 32);

        const float mk    = mask[grow] ? 1.f : 0.f;
        const float dist  = sqrtf(ss) * mk;                  // masked magnitude
        const float scale = dist / fmaxf(dist, EPSF);        // dist * dir, dir = diff/max(dist,eps)

        float2 o;
        o.x = fmaf(scale, d0, xv.x);
        o.y = fmaf(scale, d1, xv.y);
        ((float2*)out_q)[grow * 32 + lane] = o;
        if (lane == 0) out_idx[grow] = (float)idx;
    }
}

// ---------------------------------------------------------------------------
extern "C" void kernel_launch(void* const* d_in, const int* in_sizes, int n_in,
                              void* d_out, int out_size, void* d_ws, size_t ws_size,
                              hipStream_t stream)
{
    const float*         x      = (const float*)d_in[0];          // 8*4096*64
    const unsigned char* mask   = (const unsigned char*)d_in[1];  // 8*4096 bool
    const float*         frozen = (const float*)d_in[2];          // 8192*64
    const float*         W      = (const float*)d_in[3];          // 64*64

    float* out      = (float*)d_out;
    float* out_q    = out;                      // 2097152
    float* out_idx  = out + NROWS * DIMD;       // 32768
    float* out_loss = out + NROWS * DIMD + NROWS; // 1

    float2* pk    = (float2*)d_ws;                                   // 2 MB
    float*  cnorm = (float*)((char*)d_ws + 32 * KCODES * sizeof(float2)); // 32 KB

    diveq_prep<<<KCODES, 64, 0, stream>>>(frozen, W, pk, cnorm, out_loss);
    diveq_main<<<NROWS / 128, 256, 0, stream>>>(x, mask, pk, cnorm, out_q, out_idx);
}